// CRF_30751965839484
// MI455X (gfx1250) — compile-verified
//
#include <hip/hip_runtime.h>
#include <stdint.h>

#define HH 512
#define WW 512
#define NC 21          // channels
#define CP 24          // padded channels (96B rows -> 6 x float4)
#define NPIX (HH*WW)
#define D1B 6          // bilateral d+1
#define D1S 3          // spatial d+1
#define NITER 5
#define ALPHA_B (32.0f/33.0f)   // 1/(1+2^-(6-1))
#define ALPHA_S (0.8f)          // 1/(1+2^-(3-1))

typedef float v2f  __attribute__((ext_vector_type(2)));
typedef float v8f  __attribute__((ext_vector_type(8)));
typedef unsigned int u32x4 __attribute__((ext_vector_type(4)));
typedef int   i32x8 __attribute__((ext_vector_type(8)));
typedef int   i32x4 __attribute__((ext_vector_type(4)));

// ---------------------------------------------------------------- utilities

__global__ void zero_kernel(float* __restrict__ p, long n) {
  long i = (long)blockIdx.x * blockDim.x + threadIdx.x;
  if (i < n) p[i] = 0.0f;
}

// Copy padded Q (N,24) -> output (N,21)
__global__ void writeout_kernel(const float* __restrict__ Qb, float* __restrict__ out) {
  long i = (long)blockIdx.x * blockDim.x + threadIdx.x;
  if (i >= (long)NPIX * NC) return;
  long p = i / NC;
  int  c = (int)(i % NC);
  out[i] = Qb[p * CP + c];
}

// ---------------------------------------------------------------- splat
// one thread per (pixel, vertex). nch==1 -> splat of ones (normalization).
__global__ void splat_kernel(const float* __restrict__ q,
                             const float* __restrict__ ws,
                             const int*   __restrict__ os,
                             float* __restrict__ vals,
                             int d1, int nch, int stride, long total)
{
  long i = (long)blockIdx.x * blockDim.x + threadIdx.x;
  if (i >= total) return;
  int p = (int)(i / d1);
  int k = (int)(i % d1);
  float w = ws[(long)p * d1 + k];
  int   m = os[(long)p * d1 + k] + 1;      // +1: row 0 is the zero boundary row
  float* dst = vals + (long)m * stride;
  if (nch == 1) {
    atomicAdd(dst, w);
  } else {
    const float* src = q + (long)p * CP;
#pragma unroll
    for (int c = 0; c < NC; ++c) atomicAdd(dst + c, w * src[c]);
  }
}

// ---------------------------------------------------------------- TDM helper
// 1-D contiguous DMA: nelem fp32 words, global -> LDS. Descriptor per ISA ch.8:
// group0: count=1, lds_addr, global_addr[56:0], type=2
// group1: data_size=4B, tensor_dim0=tile_dim0=nelem (1-D tile), dim0 stride=nelem
// groups 2/3 (higher dims) and trailing group zeroed.
__device__ __forceinline__ void tdm_load_rows(const float* gsrc, float* ldst, int nelem)
{
  uint64_t ga = (uint64_t)(uintptr_t)gsrc;
  uint32_t la = (uint32_t)(uintptr_t)ldst;   // low 32 bits of generic ptr = LDS byte offset
  u32x4 g0;
  g0.x = 1u;                                             // count=1, no gather/restore
  g0.y = la;                                             // lds_addr
  g0.z = (uint32_t)ga;                                   // global_addr[31:0]
  g0.w = (uint32_t)((ga >> 32) & 0x01FFFFFFu) | (2u << 30); // global_addr[56:32] | type=2
  i32x8 g1;
  uint32_t ne = (uint32_t)nelem;
  g1[0] = (int)(2u << 16);                               // wg_mask=0, data_size=2 (4 bytes)
  g1[1] = (int)((ne & 0xFFFFu) << 16);                   // tensor_dim0[15:0] @ bits63:48
  g1[2] = (int)(((ne >> 16) & 0xFFFFu) | (1u << 16));    // tensor_dim0[31:16] | tensor_dim1=1
  g1[3] = (int)((ne & 0xFFFFu) << 16);                   // tile_dim0 @ bits127:112
  g1[4] = 0;                                             // tile_dim1=0, tile_dim2=0 (1-D)
  g1[5] = (int)ne;                                       // tensor_dim0_stride[31:0]
  g1[6] = 0;
  g1[7] = 0;
  i32x4 g2 = {0, 0, 0, 0};
  i32x4 g3 = {0, 0, 0, 0};
  i32x8 g4 = {0, 0, 0, 0, 0, 0, 0, 0};
  __builtin_amdgcn_tensor_load_to_lds(g0, g1, g2, g3, g4, 0);
}

// ---------------------------------------------------------------- blur (C=24)
// out[m+1] = in[m+1] + 0.5*(in[n1] + in[n2]); center rows DMA-staged via TDM.
#define BTILE 256
__launch_bounds__(256)
__global__ void blur24_kernel(const float* __restrict__ vin, float* __restrict__ vout,
                              const int* __restrict__ bn, int M, int jdir)
{
  __shared__ __align__(16) float tile[BTILE * CP];     // 24 KB of 320 KB WGP LDS
  const int m0   = blockIdx.x * BTILE;
  const int rows = (M - m0 < BTILE) ? (M - m0) : BTILE;

  if (threadIdx.x == 0) {                               // one TDM issue per block
    tdm_load_rows(vin + (long)(m0 + 1) * CP, tile, rows * CP);
    __builtin_amdgcn_s_wait_tensorcnt(0);
  }
  __syncthreads();

  if (threadIdx.x < rows) {
    const int m = m0 + threadIdx.x;
    const long bb = ((long)jdir * M + m) * 2;
    const int n1 = bn[bb + 0];                          // indices already +1 shifted; 0 = zero row
    const int n2 = bn[bb + 1];
    const float4* r1 = (const float4*)(vin + (long)n1 * CP);
    const float4* r2 = (const float4*)(vin + (long)n2 * CP);
    const float4* rc = (const float4*)(tile + threadIdx.x * CP);
    float4* out = (float4*)(vout + (long)(m + 1) * CP);
#pragma unroll
    for (int q4 = 0; q4 < 6; ++q4) {
      float4 a = rc[q4], b = r1[q4], c = r2[q4];
      out[q4] = make_float4(a.x + 0.5f * (b.x + c.x),
                            a.y + 0.5f * (b.y + c.y),
                            a.z + 0.5f * (b.z + c.z),
                            a.w + 0.5f * (b.w + c.w));
    }
  }
}

// ---------------------------------------------------------------- blur (C=1)
__global__ void blur1_kernel(const float* __restrict__ vin, float* __restrict__ vout,
                             const int* __restrict__ bn, int M, int jdir)
{
  int m = blockIdx.x * blockDim.x + threadIdx.x;
  if (m >= M) return;
  long bb = ((long)jdir * M + m) * 2;
  vout[m + 1] = vin[m + 1] + 0.5f * (vin[bn[bb]] + vin[bn[bb + 1]]);
}

// ---------------------------------------------------------------- norm slice
__global__ void slice_norm_kernel(const float* __restrict__ vals,
                                  const float* __restrict__ ws,
                                  const int*   __restrict__ os,
                                  float* __restrict__ invout,
                                  int d1, float alpha)
{
  int p = blockIdx.x * blockDim.x + threadIdx.x;
  if (p >= NPIX) return;
  float s = 0.f;
  for (int k = 0; k < d1; ++k)
    s += ws[(long)p * d1 + k] * vals[os[(long)p * d1 + k] + 1];
  invout[p] = 1.0f / (alpha * s + 1e-20f);
}

// ---------------------------------------------------------------- update
// logits = -u + 10*b + 3*s ; Q = softmax(logits). Softmax denominators computed
// with exact fp32 WMMA 16x16x4 (E-tile x ones), per ISA A/C fragment layouts.
__launch_bounds__(256)
__global__ void update_kernel(const float* __restrict__ unary,
                              float* __restrict__ Qb,
                              const float* __restrict__ vB, const float* __restrict__ wsB,
                              const int*   __restrict__ osB, const float* __restrict__ invNb,
                              const float* __restrict__ vS, const float* __restrict__ wsS,
                              const int*   __restrict__ osS, const float* __restrict__ invNs,
                              int use_pairwise)
{
  __shared__ __align__(16) float E[256 * CP];   // exp tile, pads zero
  __shared__ float denom[256];
  const int t = threadIdx.x;
  const int p = blockIdx.x * 256 + t;           // NPIX % 256 == 0 -> EXEC all ones for WMMA

  float logit[NC];
  {
    const float* u = unary + (long)p * NC;
#pragma unroll
    for (int c = 0; c < NC; ++c) logit[c] = -u[c];
  }

  if (use_pairwise) {
    float acc[CP];
    // bilateral slice
#pragma unroll
    for (int c = 0; c < CP; ++c) acc[c] = 0.f;
    for (int k = 0; k < D1B; ++k) {
      float w = wsB[(long)p * D1B + k];
      int   m = osB[(long)p * D1B + k] + 1;
      const float4* row = (const float4*)(vB + (long)m * CP);
#pragma unroll
      for (int q4 = 0; q4 < 6; ++q4) {
        float4 r = row[q4];
        acc[q4*4+0] += w * r.x; acc[q4*4+1] += w * r.y;
        acc[q4*4+2] += w * r.z; acc[q4*4+3] += w * r.w;
      }
    }
    float sc = 10.0f * ALPHA_B * invNb[p];
#pragma unroll
    for (int c = 0; c < NC; ++c) logit[c] += sc * acc[c];
    // spatial slice
#pragma unroll
    for (int c = 0; c < CP; ++c) acc[c] = 0.f;
    for (int k = 0; k < D1S; ++k) {
      float w = wsS[(long)p * D1S + k];
      int   m = osS[(long)p * D1S + k] + 1;
      const float4* row = (const float4*)(vS + (long)m * CP);
#pragma unroll
      for (int q4 = 0; q4 < 6; ++q4) {
        float4 r = row[q4];
        acc[q4*4+0] += w * r.x; acc[q4*4+1] += w * r.y;
        acc[q4*4+2] += w * r.z; acc[q4*4+3] += w * r.w;
      }
    }
    sc = 3.0f * ALPHA_S * invNs[p];
#pragma unroll
    for (int c = 0; c < NC; ++c) logit[c] += sc * acc[c];
  }

  float mx = logit[0];
#pragma unroll
  for (int c = 1; c < NC; ++c) mx = fmaxf(mx, logit[c]);
#pragma unroll
  for (int c = 0; c < NC; ++c) E[t * CP + c] = __expf(logit[c] - mx);
  E[t * CP + 21] = 0.f; E[t * CP + 22] = 0.f; E[t * CP + 23] = 0.f;
  __syncthreads();

  // --- WMMA row sums: D = sum_j A_j(16x4 chunk of E) * ones(4x16), exact fp32.
  const int wv = t >> 5, ln = t & 31;
  const int rbase = ln & 15;
  const int cb = (ln >> 4) * 2;   // A layout: lanes 0-15 K={0,1}, lanes 16-31 K={2,3}
  v2f bones; bones.x = 1.0f; bones.y = 1.0f;
#pragma unroll
  for (int half = 0; half < 2; ++half) {
    const int R = (wv * 2 + half) * 16;  // 8 waves x 2 tiles = 256 rows
    const int row = R + rbase;
    v8f d = {0.f, 0.f, 0.f, 0.f, 0.f, 0.f, 0.f, 0.f};
#pragma unroll
    for (int j = 0; j < 6; ++j) {
      v2f a;
      a.x = E[row * CP + 4 * j + cb + 0];
      a.y = E[row * CP + 4 * j + cb + 1];
      d = __builtin_amdgcn_wmma_f32_16x16x4_f32(false, a, false, bones,
                                                (short)0, d, false, false);
    }
    // C/D layout: lane 0 col0 holds rows 0..7 (VGPR v = row v); lane 16 rows 8..15
    if (ln == 0) {
#pragma unroll
      for (int v = 0; v < 8; ++v) denom[R + v] = d[v];
    } else if (ln == 16) {
#pragma unroll
      for (int v = 0; v < 8; ++v) denom[R + 8 + v] = d[v];
    }
  }
  __syncthreads();

  const float inv = 1.0f / denom[t];
  float4* qo = (float4*)(Qb + (long)p * CP);
  const float4* es = (const float4*)(E + t * CP);
#pragma unroll
  for (int q4 = 0; q4 < 6; ++q4) {
    float4 e = es[q4];
    qo[q4] = make_float4(e.x * inv, e.y * inv, e.z * inv, e.w * inv);
  }
}

// ---------------------------------------------------------------- host driver

static inline size_t alignup256(size_t x) { return (x + 255) & ~(size_t)255; }
static inline unsigned nblk(long n, int tb) { return (unsigned)((n + tb - 1) / tb); }

extern "C" void kernel_launch(void* const* d_in, const int* in_sizes, int n_in,
                              void* d_out, int out_size, void* d_ws, size_t ws_size,
                              hipStream_t stream)
{
  const float* unary = (const float*)d_in[0];
  const float* wsB   = (const float*)d_in[1];
  const int*   osB   = (const int*)  d_in[2];
  const int*   bnB   = (const int*)  d_in[3];
  const float* wsS   = (const float*)d_in[5];
  const int*   osS   = (const int*)  d_in[6];
  const int*   bnS   = (const int*)  d_in[7];
  const int M_b = in_sizes[3] / (D1B * 2);   // bn_bilateral: (6, M, 2)
  const int M_s = in_sizes[7] / (D1S * 2);   // bn_spatial:   (3, M, 2)
  (void)n_in; (void)ws_size; (void)out_size;

  char* w = (char*)d_ws;
  float* Qb    = (float*)w; w += alignup256((size_t)NPIX * CP * 4);
  float* invNb = (float*)w; w += alignup256((size_t)NPIX * 4);
  float* invNs = (float*)w; w += alignup256((size_t)NPIX * 4);
  float* vB0   = (float*)w; w += alignup256((size_t)(M_b + 1) * CP * 4);
  float* vB1   = (float*)w; w += alignup256((size_t)(M_b + 1) * CP * 4);
  float* vS0   = (float*)w; w += alignup256((size_t)(M_s + 1) * CP * 4);
  float* vS1   = (float*)w; w += alignup256((size_t)(M_s + 1) * CP * 4);

  const int TB = 256;
  const long nvB = (long)(M_b + 1) * CP;
  const long nvS = (long)(M_s + 1) * CP;

  // Q0 = softmax(-u)
  update_kernel<<<NPIX / TB, TB, 0, stream>>>(unary, Qb,
      nullptr, nullptr, nullptr, nullptr, nullptr, nullptr, nullptr, nullptr, 0);

  // ---- normalization: inv_nb = 1/filter_b(1), inv_ns = 1/filter_s(1)
  {
    zero_kernel<<<nblk(nvB, TB), TB, 0, stream>>>(vB0, nvB);
    zero_kernel<<<nblk(nvB, TB), TB, 0, stream>>>(vB1, nvB);
    splat_kernel<<<nblk((long)NPIX * D1B, TB), TB, 0, stream>>>(
        nullptr, wsB, osB, vB0, D1B, 1, 1, (long)NPIX * D1B);
    float *pin = vB0, *pout = vB1;
    for (int j = 0; j < D1B; ++j) {
      blur1_kernel<<<nblk(M_b, TB), TB, 0, stream>>>(pin, pout, bnB, M_b, j);
      float* tmp = pin; pin = pout; pout = tmp;
    }
    slice_norm_kernel<<<nblk(NPIX, TB), TB, 0, stream>>>(pin, wsB, osB, invNb, D1B, ALPHA_B);
  }
  {
    zero_kernel<<<nblk(nvS, TB), TB, 0, stream>>>(vS0, nvS);
    zero_kernel<<<nblk(nvS, TB), TB, 0, stream>>>(vS1, nvS);
    splat_kernel<<<nblk((long)NPIX * D1S, TB), TB, 0, stream>>>(
        nullptr, wsS, osS, vS0, D1S, 1, 1, (long)NPIX * D1S);
    float *pin = vS0, *pout = vS1;
    for (int j = 0; j < D1S; ++j) {
      blur1_kernel<<<nblk(M_s, TB), TB, 0, stream>>>(pin, pout, bnS, M_s, j);
      float* tmp = pin; pin = pout; pout = tmp;
    }
    slice_norm_kernel<<<nblk(NPIX, TB), TB, 0, stream>>>(pin, wsS, osS, invNs, D1S, ALPHA_S);
  }

  // ---- 5 mean-field iterations
  for (int it = 0; it < NITER; ++it) {
    // bilateral filter of Q
    zero_kernel<<<nblk(nvB, TB), TB, 0, stream>>>(vB0, nvB);
    zero_kernel<<<nblk(nvB, TB), TB, 0, stream>>>(vB1, nvB);
    splat_kernel<<<nblk((long)NPIX * D1B, TB), TB, 0, stream>>>(
        Qb, wsB, osB, vB0, D1B, NC, CP, (long)NPIX * D1B);
    float *pinB = vB0, *poutB = vB1;
    for (int j = 0; j < D1B; ++j) {
      blur24_kernel<<<nblk(M_b, BTILE), BTILE, 0, stream>>>(pinB, poutB, bnB, M_b, j);
      float* tmp = pinB; pinB = poutB; poutB = tmp;
    }
    // spatial filter of Q
    zero_kernel<<<nblk(nvS, TB), TB, 0, stream>>>(vS0, nvS);
    zero_kernel<<<nblk(nvS, TB), TB, 0, stream>>>(vS1, nvS);
    splat_kernel<<<nblk((long)NPIX * D1S, TB), TB, 0, stream>>>(
        Qb, wsS, osS, vS0, D1S, NC, CP, (long)NPIX * D1S);
    float *pinS = vS0, *poutS = vS1;
    for (int j = 0; j < D1S; ++j) {
      blur24_kernel<<<nblk(M_s, BTILE), BTILE, 0, stream>>>(pinS, poutS, bnS, M_s, j);
      float* tmp = pinS; pinS = poutS; poutS = tmp;
    }
    // slice + pairwise + softmax -> new Q
    update_kernel<<<NPIX / TB, TB, 0, stream>>>(unary, Qb,
        pinB, wsB, osB, invNb, pinS, wsS, osS, invNs, 1);
  }

  writeout_kernel<<<nblk((long)NPIX * NC, TB), TB, 0, stream>>>(Qb, (float*)d_out);
}